// _LargeKernelAttention_89026082111869
// MI455X (gfx1250) — compile-verified
//
#include <hip/hip_runtime.h>

typedef __attribute__((ext_vector_type(2))) float v2f;
typedef __attribute__((ext_vector_type(4))) float v4f;
typedef __attribute__((ext_vector_type(8))) float v8f;

#define PSTR  100                 // padded LDS row stride (floats), 16B-aligned rows
#define POFF  16                  // halo offset (rows/cols -16..80 valid)
#define PROWS 97
#define PFLOATS (PROWS * PSTR)    // 9700 floats per plane
#define PIDX(r, c) (((r) + POFF) * PSTR + (c) + POFF)

__global__ __launch_bounds__(256) void lka_plane_kernel(
    const float* __restrict__ x,
    const float* __restrict__ w0,  const float* __restrict__ b0,
    const float* __restrict__ w01, const float* __restrict__ b01,
    const float* __restrict__ w02, const float* __restrict__ b02,
    const float* __restrict__ w11, const float* __restrict__ b11,
    const float* __restrict__ w12, const float* __restrict__ b12,
    const float* __restrict__ w21, const float* __restrict__ b21,
    const float* __restrict__ w22, const float* __restrict__ b22,
    float* __restrict__ out)
{
    __shared__ __align__(16) float s_x[PFLOATS];     // x plane (halo = 0); reused as store staging
    __shared__ __align__(16) float s_attn[PFLOATS];  // 5x5 result (halo = 0)
    __shared__ __align__(16) float s_tmp[PFLOATS];   // horizontal intermediate (halo = 0)
    __shared__ __align__(16) float s_w[704];         // zero-padded Toeplitz weight sources

    const int tid  = threadIdx.x;
    const int lane = tid & 31;
    const int wv   = tid >> 5;     // wave id 0..7
    const int half = lane >> 4;    // selects K pair within a chunk
    const int ln   = lane & 15;    // M (A) or N (B/D) index

    const int bc = blockIdx.x;     // b*128 + c
    const int b  = bc >> 7;
    const int c  = bc & 127;

    // ---- zero planes (establishes zero halos once) + weight pad area ----
    {
        const v4f z4 = {0.f, 0.f, 0.f, 0.f};
        for (int q = tid; q < PFLOATS / 4; q += 256) {
            ((v4f*)s_x)[q]    = z4;
            ((v4f*)s_attn)[q] = z4;
            ((v4f*)s_tmp)[q]  = z4;
        }
        for (int i = tid; i < 704; i += 256) s_w[i] = 0.0f;
    }
    __syncthreads();

    // ---- stage per-channel weights into zero-padded LDS arrays ----
    if (tid < 25) s_w[(tid / 5) * 40 + 16 + (tid % 5)] = w0[c * 25 + tid]; // 5 rows of 5x5
    if (tid < 7)  s_w[200 + 0 * 80 + 16 + tid] = w01[c * 7  + tid];        // horizontal 1xK
    if (tid < 15) s_w[200 + 1 * 80 + 16 + tid] = w11[c * 15 + tid];
    if (tid < 31) s_w[200 + 2 * 80 + 16 + tid] = w21[c * 31 + tid];
    if (tid < 7)  s_w[440 + 0 * 80 + 16 + tid] = w02[c * 7  + tid];        // vertical Kx1
    if (tid < 15) s_w[440 + 1 * 80 + 16 + tid] = w12[c * 15 + tid];
    if (tid < 31) s_w[440 + 2 * 80 + 16 + tid] = w22[c * 31 + tid];

    // ---- load x plane into LDS interior; stream-copy to out group 0 (NT: read/write-once) ----
    const float* xp      = x   + (size_t)(b * 128 + c) * 4096;
    float*       outbase = out + (size_t)(b * 640 + c) * 4096;
    for (int q = tid; q < 1024; q += 256) {
        int row = q >> 4, c4 = (q & 15) << 2;
        v4f v = __builtin_nontemporal_load((const v4f*)(xp + row * 64 + c4));
        *(v4f*)(&s_x[PIDX(row, c4)]) = v;
        __builtin_nontemporal_store(v, (v4f*)(outbase + row * 64 + c4));
    }
    __syncthreads();

    const float bias0 = b0[c];

    // ---- attn = dwconv5x5(x)+b0 : 5 vertical taps x banded-Toeplitz WMMA over W ----
    {
        // hoist all 25 Toeplitz B fragments (shared by both tiles of this wave)
        v2f bf[25];
        #pragma unroll
        for (int r = 0; r < 5; ++r) {
            #pragma unroll
            for (int ch = 0; ch < 5; ++ch) {
                const int k0 = ch * 4 + 2 * half;
                bf[r * 5 + ch].x = s_w[r * 40 + 16 + k0 - ln];      // T[i,n] = w5[r][i-n]
                bf[r * 5 + ch].y = s_w[r * 40 + 16 + k0 + 1 - ln];
            }
        }
        for (int t = wv; t < 16; t += 8) {
            const int h0 = (t >> 2) << 4, wc = (t & 3) << 4;
            v8f acc;
            #pragma unroll
            for (int j = 0; j < 8; ++j) acc[j] = bias0;
            #pragma unroll
            for (int r = 0; r < 5; ++r) {
                const int row = h0 + ln + r - 2;                    // halo makes this always safe
                #pragma unroll
                for (int ch = 0; ch < 5; ++ch) {
                    const int k0 = ch * 4 + 2 * half;
                    const int c0 = wc - 2 + k0;
                    v2f a;
                    a.x = s_x[PIDX(row, c0)];
                    a.y = s_x[PIDX(row, c0 + 1)];
                    acc = __builtin_amdgcn_wmma_f32_16x16x4_f32(
                              false, a, false, bf[r * 5 + ch], (short)0, acc, false, false);
                }
            }
            #pragma unroll
            for (int j = 0; j < 8; ++j)
                s_attn[PIDX(h0 + j + 8 * half, wc + ln)] = acc[j];
        }
    }
    __syncthreads();

    // attn -> out group 1
    {
        float* g = outbase + (size_t)128 * 4096;
        for (int q = tid; q < 1024; q += 256) {
            int row = q >> 4, c4 = (q & 15) << 2;
            v4f v = *(const v4f*)(&s_attn[PIDX(row, c4)]);
            __builtin_nontemporal_store(v, (v4f*)(g + row * 64 + c4));
        }
    }

    // ---- three separable branches: 1xK then Kx1, K in {7,15,31} ----
    const int Ps[3] = {3, 7, 15};
    const int NCs[3] = {6, 8, 12};             // ceil((K+15)/4) chunks
    const float bh[3]  = {b01[c], b11[c], b21[c]};
    const float bvv[3] = {b02[c], b12[c], b22[c]};

    #pragma unroll
    for (int br = 0; br < 3; ++br) {
        const int P = Ps[br], NC = NCs[br];
        const float* wh  = &s_w[200 + br * 80 + 16];
        const float* wvp = &s_w[440 + br * 80 + 16];

        // hoisted Toeplitz fragments for this branch (reused across both tiles)
        v2f bfh[12], afv[12];
        #pragma unroll
        for (int ch = 0; ch < 12; ++ch) {
            if (ch < NC) {
                const int k0 = ch * 4 + 2 * half;
                bfh[ch].x = wh[k0 - ln];       // horizontal B: T[i,n] = wh[i-n]
                bfh[ch].y = wh[k0 + 1 - ln];
                afv[ch].x = wvp[k0 - ln];      // vertical   A: T[m,i] = wv[i-m]
                afv[ch].y = wvp[k0 + 1 - ln];
            }
        }

        // horizontal 1xK: s_attn -> s_tmp   (A = image patch, B = Toeplitz weights)
        for (int t = wv; t < 16; t += 8) {
            const int h0 = (t >> 2) << 4, wc = (t & 3) << 4;
            const int row = h0 + ln;
            v8f acc;
            #pragma unroll
            for (int j = 0; j < 8; ++j) acc[j] = bh[br];
            #pragma unroll
            for (int ch = 0; ch < 12; ++ch) {
                if (ch < NC) {
                    const int k0 = ch * 4 + 2 * half;
                    const int c0 = wc - P + k0;
                    v2f a;
                    a.x = s_attn[PIDX(row, c0)];
                    a.y = s_attn[PIDX(row, c0 + 1)];
                    acc = __builtin_amdgcn_wmma_f32_16x16x4_f32(
                              false, a, false, bfh[ch], (short)0, acc, false, false);
                }
            }
            #pragma unroll
            for (int j = 0; j < 8; ++j)
                s_tmp[PIDX(h0 + j + 8 * half, wc + ln)] = acc[j];
        }
        __syncthreads();

        // vertical Kx1: s_tmp -> s_x staging  (A = Toeplitz weights, B = image rows)
        for (int t = wv; t < 16; t += 8) {
            const int h0 = (t >> 2) << 4, wc = (t & 3) << 4;
            v8f acc;
            #pragma unroll
            for (int j = 0; j < 8; ++j) acc[j] = bvv[br];
            #pragma unroll
            for (int ch = 0; ch < 12; ++ch) {
                if (ch < NC) {
                    const int k0 = ch * 4 + 2 * half;
                    const int r0 = h0 - P + k0;
                    v2f bb;
                    bb.x = s_tmp[PIDX(r0,     wc + ln)];
                    bb.y = s_tmp[PIDX(r0 + 1, wc + ln)];
                    acc = __builtin_amdgcn_wmma_f32_16x16x4_f32(
                              false, afv[ch], false, bb, (short)0, acc, false, false);
                }
            }
            #pragma unroll
            for (int j = 0; j < 8; ++j)
                s_x[PIDX(h0 + j + 8 * half, wc + ln)] = acc[j];
        }
        __syncthreads();

        // staging -> out group (2+br), fully coalesced NT stores
        {
            float* g = outbase + (size_t)(128 * (2 + br)) * 4096;
            for (int q = tid; q < 1024; q += 256) {
                int row = q >> 4, c4 = (q & 15) << 2;
                v4f v = *(const v4f*)(&s_x[PIDX(row, c4)]);
                __builtin_nontemporal_store(v, (v4f*)(g + row * 64 + c4));
            }
        }
        __syncthreads();   // protect s_x staging before next branch overwrites it
    }
}

extern "C" void kernel_launch(void* const* d_in, const int* in_sizes, int n_in,
                              void* d_out, int out_size, void* d_ws, size_t ws_size,
                              hipStream_t stream) {
    const float* x   = (const float*)d_in[0];
    const float* w0  = (const float*)d_in[1];
    const float* b0  = (const float*)d_in[2];
    const float* w01 = (const float*)d_in[3];
    const float* b01 = (const float*)d_in[4];
    const float* w02 = (const float*)d_in[5];
    const float* b02 = (const float*)d_in[6];
    const float* w11 = (const float*)d_in[7];
    const float* b11 = (const float*)d_in[8];
    const float* w12 = (const float*)d_in[9];
    const float* b12 = (const float*)d_in[10];
    const float* w21 = (const float*)d_in[11];
    const float* b21 = (const float*)d_in[12];
    const float* w22 = (const float*)d_in[13];
    const float* b22 = (const float*)d_in[14];
    float* out = (float*)d_out;

    dim3 grid(32 * 128);   // one workgroup per (batch, channel) plane
    dim3 block(256);       // 8 wave32s
    hipLaunchKernelGGL(lka_plane_kernel, grid, block, 0, stream,
                       x, w0, b0, w01, b01, w02, b02,
                       w11, b11, w12, b12, w21, b21, w22, b22, out);
}